// ModelNew_68161130987849
// MI455X (gfx1250) — compile-verified
//
#include <hip/hip_runtime.h>

// fp32 WMMA tile GEMM for C = triu(triu(A) @ triu(B)) on gfx1250 (wave32).
// Block tile 128x128, K-chunk 32, 256 threads = 8 waves, each wave = 64x32 of C
// via 4x2 V_WMMA_F32_16X16X4_F32 accumulators.
//  - k-loop restricted per block to [bi*128, (bj+1)*128)  (triangular structure)
//  - global->reg->LDS pipelining: next chunk's HBM loads issue before compute
//  - fragment ping-pong: LDS loads for K-step s+1 issue before WMMAs of step s
//  - triu masks applied only on chunks that touch the diagonal (uniform branch)

typedef __attribute__((ext_vector_type(2))) float v2f;
typedef __attribute__((ext_vector_type(4))) float v4f;
typedef __attribute__((ext_vector_type(8))) float v8f;

#define TRIU_BM 128
#define TRIU_BN 128
#define TRIU_BK 32
#define TRIU_LDA 36    // BK + 4 pad: A fragment reads conflict-free
#define TRIU_LDB 136   // BN + 8 pad: half-wave B rows land 16 banks apart

__global__ __launch_bounds__(256, 2)
void triu_gemm_wmma_f32(const float* __restrict__ A,
                        const float* __restrict__ B,
                        float* __restrict__ C, int N)
{
    const int bj = blockIdx.x;            // column block of C
    const int bi = blockIdx.y;            // row block of C
    const int rowStart = bi * TRIU_BM;
    const int colStart = bj * TRIU_BN;
    const int tid  = threadIdx.x;
    const int lane = tid & 31;
    const int wave = tid >> 5;            // 0..7
    const int wm   = wave >> 2;           // 0..1 : 64-row slab
    const int wn   = wave & 3;            // 0..3 : 32-col slab

    // Blocks strictly below the diagonal are all-zero: just store zeros.
    if (bi > bj) {
        v4f z = {};
        #pragma unroll
        for (int it = 0; it < 16; ++it) {
            int v = tid + 256 * it;
            int r = v >> 5;
            int c = (v & 31) << 2;
            *(v4f*)(C + (size_t)(rowStart + r) * N + (colStart + c)) = z;
        }
        return;
    }

    __shared__ float As[TRIU_BM][TRIU_LDA];
    __shared__ float Bs[TRIU_BK][TRIU_LDB];

    v8f acc[4][2];
    {
        v8f zero = {};
        #pragma unroll
        for (int tm = 0; tm < 4; ++tm)
            #pragma unroll
            for (int tn = 0; tn < 2; ++tn)
                acc[tm][tn] = zero;
    }

    const int kBeg = rowStart;               // multiple of BK
    const int kEnd = colStart + TRIU_BN;     // multiple of BK, <= N

    const int m16 = lane & 15;               // M (A) / N (B,C) index
    const int kh  = lane >> 4;               // K half-select

    // Per-thread staging coordinates (constant across chunks).
    const int arL[4] = { (tid + 0) >> 3, (tid + 256) >> 3,
                         (tid + 512) >> 3, (tid + 768) >> 3 };
    const int acL    = (tid & 7) << 2;       // A col within chunk
    const int brL[4] = { (tid + 0) >> 5, (tid + 256) >> 5,
                         (tid + 512) >> 5, (tid + 768) >> 5 };
    const int bcL    = (tid & 31) << 2;      // B col within tile

    v4f ra[4], rb[4];                        // register staging buffers

    // ---- chunk loader: HBM -> registers, with diagonal masking as needed ----
    auto loadChunk = [&](int kc) {
        const bool maskA = (kc < rowStart + TRIU_BM);          // chunk hits A's diagonal band
        const bool maskB = (kc + TRIU_BK - 1 > colStart);      // chunk hits B's diagonal band
        #pragma unroll
        for (int it = 0; it < 4; ++it) {
            int gi = rowStart + arL[it];
            int gk = kc + acL;
            v4f va = *(const v4f*)(A + (size_t)gi * N + gk);
            if (maskA) {                      // A[i][k] used only when k >= i
                va.x = (gk + 0 >= gi) ? va.x : 0.0f;
                va.y = (gk + 1 >= gi) ? va.y : 0.0f;
                va.z = (gk + 2 >= gi) ? va.z : 0.0f;
                va.w = (gk + 3 >= gi) ? va.w : 0.0f;
            }
            ra[it] = va;
        }
        #pragma unroll
        for (int it = 0; it < 4; ++it) {
            int gk = kc + brL[it];
            int gj = colStart + bcL;
            v4f vb = *(const v4f*)(B + (size_t)gk * N + gj);
            if (maskB) {                      // B[k][j] used only when j >= k
                vb.x = (gj + 0 >= gk) ? vb.x : 0.0f;
                vb.y = (gj + 1 >= gk) ? vb.y : 0.0f;
                vb.z = (gj + 2 >= gk) ? vb.z : 0.0f;
                vb.w = (gj + 3 >= gk) ? vb.w : 0.0f;
            }
            rb[it] = vb;
        }
    };

    // ---- fragment loader: LDS -> VGPR frags for one K-step ----
    auto loadFrag = [&](int kk, v2f af[4], v2f bf[2]) {
        #pragma unroll
        for (int tm = 0; tm < 4; ++tm)
            af[tm] = *(const v2f*)&As[wm * 64 + tm * 16 + m16][kk + 2 * kh];
        #pragma unroll
        for (int tn = 0; tn < 2; ++tn) {
            int col = wn * 32 + tn * 16 + m16;
            bf[tn].x = Bs[kk + 2 * kh + 0][col];
            bf[tn].y = Bs[kk + 2 * kh + 1][col];
        }
    };

    loadChunk(kBeg);                          // prologue: first chunk into regs

    for (int kc = kBeg; kc < kEnd; kc += TRIU_BK) {
        __syncthreads();                      // previous chunk's LDS reads done
        #pragma unroll
        for (int it = 0; it < 4; ++it)
            *(v4f*)&As[arL[it]][acL] = ra[it];
        #pragma unroll
        for (int it = 0; it < 4; ++it)
            *(v4f*)&Bs[brL[it]][bcL] = rb[it];
        __syncthreads();                      // tile visible to all waves

        if (kc + TRIU_BK < kEnd)              // issue next chunk's HBM loads now;
            loadChunk(kc + TRIU_BK);          // latency hides under 64 WMMAs below

        // 8 K-steps of 4, fragment ping-pong => ds loads lead WMMAs by a step.
        v2f af[2][4];
        v2f bf[2][2];
        loadFrag(0, af[0], bf[0]);
        #pragma unroll
        for (int s = 0; s < 8; ++s) {
            const int cur = s & 1, nxt = cur ^ 1;
            if (s < 7)
                loadFrag((s + 1) * 4, af[nxt], bf[nxt]);
            #pragma unroll
            for (int tm = 0; tm < 4; ++tm)
                #pragma unroll
                for (int tn = 0; tn < 2; ++tn)
                    acc[tm][tn] = __builtin_amdgcn_wmma_f32_16x16x4_f32(
                        false, af[cur][tm], false, bf[cur][tn],
                        (short)0, acc[tm][tn], false, false);
        }
    }

    // Store, applying the output triu mask (cuts through diagonal blocks).
    #pragma unroll
    for (int tm = 0; tm < 4; ++tm) {
        #pragma unroll
        for (int tn = 0; tn < 2; ++tn) {
            #pragma unroll
            for (int r = 0; r < 8; ++r) {
                int grow = rowStart + wm * 64 + tm * 16 + r + 8 * kh;
                int gcol = colStart + wn * 32 + tn * 16 + m16;
                float val = acc[tm][tn][r];
                C[(size_t)grow * N + gcol] = (grow <= gcol) ? val : 0.0f;
            }
        }
    }
}

extern "C" void kernel_launch(void* const* d_in, const int* in_sizes, int n_in,
                              void* d_out, int out_size, void* d_ws, size_t ws_size,
                              hipStream_t stream) {
    const float* A = (const float*)d_in[0];
    const float* B = (const float*)d_in[1];
    float* C = (float*)d_out;

    int N = 4096;  // reference fixes N=4096; derive defensively from out_size
    {
        long long t = 1;
        while (t * t < (long long)out_size) ++t;
        if (t * t == (long long)out_size) N = (int)t;
    }

    dim3 grid(N / TRIU_BN, N / TRIU_BM);
    triu_gemm_wmma_f32<<<grid, dim3(256), 0, stream>>>(A, B, C, N);
}